// PPO_69337952026821
// MI455X (gfx1250) — compile-verified
//
#include <hip/hip_runtime.h>
#include <cstdint>

#define GAMMA 0.99f
#define GTAU  (0.99f * 0.95f)

constexpr int BT     = 256;        // threads per block
constexpr int T      = 16;         // elements per thread (contiguous segment)
constexpr int CHUNK  = BT * T;     // 4096 elements per block
constexpr int TP     = T + 2;      // padded LDS stride: 8B-aligned rows, bank-conflict free
constexpr int MAXBLK = 1024;

static_assert(T == 16, "stage_tile index math assumes T==16");

struct Aff { float aV, bV, aA, bA; };   // x_out = a*x_in + b, for vt and A recurrences

__device__ __forceinline__ Aff aff_compose(Aff L, Aff R) {
  // L covers lower t, R covers higher t (applied first in the reverse scan)
  Aff o;
  o.aV = L.aV * R.aV;
  o.bV = fmaf(L.aV, R.bV, L.bV);
  o.aA = L.aA * R.aA;
  o.bA = fmaf(L.aA, R.bA, L.bA);
  return o;
}

// ---- CDNA5 async global->LDS copies (ASYNCcnt-tracked DMA into LDS) ----
__device__ __forceinline__ void async_ld32(unsigned lds_addr, unsigned gbyte_off,
                                           const void* sbase) {
  asm volatile("global_load_async_to_lds_b32 %0, %1, %2"
               :: "v"(lds_addr), "v"(gbyte_off), "s"(sbase) : "memory");
}
__device__ __forceinline__ void async_ld64(unsigned lds_addr, unsigned gbyte_off,
                                           const void* sbase) {
  asm volatile("global_load_async_to_lds_b64 %0, %1, %2"
               :: "v"(lds_addr), "v"(gbyte_off), "s"(sbase) : "memory");
}
__device__ __forceinline__ void wait_async0() {
  asm volatile("s_wait_asynccnt 0x0" ::: "memory");
}

// Stage one chunk (r, v, mask) coalesced into padded LDS via b64 async DMA.
// Fast path: block fully in bounds (branch-free async clause).
// Tail path: per-element guard; OOB elements become the (a=0,b=0) map, which is
// exact against the x_init = 0 boundary condition of the reverse scan.
__device__ __forceinline__ void stage_tile(float* lr, float* lv, int* lm,
                                           const float* r, const float* v,
                                           const int* m, int base, int B, int tid) {
  if (base + CHUNK <= B) {                      // block-uniform fast path
#pragma unroll
    for (int i = 0; i < T / 2; ++i) {           // 8 iterations, 2 elements per lane
      int e = (i * BT + tid) * 2;               // element pair within chunk (coalesced)
      int d = (e >> 4) * TP + (e & (T - 1));    // padded per-thread-segment layout
      unsigned go = (unsigned)(base + e) * 4u;  // 8B-aligned
      async_ld64((unsigned)(uintptr_t)&lr[d], go, r);
      async_ld64((unsigned)(uintptr_t)&lv[d], go, v);
      async_ld64((unsigned)(uintptr_t)&lm[d], go, m);
    }
  } else {                                      // tail block
#pragma unroll
    for (int i = 0; i < T / 2; ++i) {
      int e = (i * BT + tid) * 2;
      int g = base + e;
      int d = (e >> 4) * TP + (e & (T - 1));
      if (g + 1 < B) {
        unsigned go = (unsigned)g * 4u;
        async_ld64((unsigned)(uintptr_t)&lr[d], go, r);
        async_ld64((unsigned)(uintptr_t)&lv[d], go, v);
        async_ld64((unsigned)(uintptr_t)&lm[d], go, m);
      } else {
#pragma unroll
        for (int q = 0; q < 2; ++q) {
          int gq = g + q, dq = d + q;
          if (gq < B) {
            unsigned go = (unsigned)gq * 4u;
            async_ld32((unsigned)(uintptr_t)&lr[dq], go, r);
            async_ld32((unsigned)(uintptr_t)&lv[dq], go, v);
            async_ld32((unsigned)(uintptr_t)&lm[dq], go, m);
          } else {
            lr[dq] = 0.0f; lv[dq] = 0.0f; lm[dq] = 0;
          }
        }
      }
    }
  }
  wait_async0();
  __syncthreads();
}

// Fold a thread's T-element segment into one affine map (both recurrences).
__device__ __forceinline__ Aff seg_summary(const float* lr, const float* lv,
                                           const int* lm, float vb, int tid) {
  Aff M = {1.0f, 0.0f, 1.0f, 0.0f};
  const int sb = tid * TP;
  float vnext = vb;
#pragma unroll
  for (int k = T - 1; k >= 0; --k) {
    float rk = lr[sb + k], vk = lv[sb + k], mk = (float)lm[sb + k];
    Aff E;
    E.aV = GAMMA * mk;
    E.bV = rk;
    E.aA = GTAU * mk;
    E.bA = rk + GAMMA * vnext * mk - vk;        // delta[t] (pointwise)
    M = aff_compose(E, M);
    vnext = vk;
  }
  return M;
}

// ---------------- K1: per-block affine summary ----------------
__global__ __launch_bounds__(BT) void k1_summary(const float* __restrict__ r,
                                                 const float* __restrict__ v,
                                                 const int* __restrict__ m,
                                                 float4* __restrict__ blocksum, int B) {
  __shared__ alignas(16) float lr[BT * TP];
  __shared__ alignas(16) float lv[BT * TP];
  __shared__ alignas(16) int   lm[BT * TP];
  __shared__ Aff               red[BT];
  const int tid  = threadIdx.x;
  const int base = blockIdx.x * CHUNK;

  stage_tile(lr, lv, lm, r, v, m, base, B, tid);

  float vb;
  if (tid < BT - 1) vb = lv[(tid + 1) * TP];
  else { int g = base + CHUNK; vb = (g < B) ? v[g] : 0.0f; }

  red[tid] = seg_summary(lr, lv, lm, vb, tid);
  __syncthreads();
#pragma unroll
  for (int s = 1; s < BT; s <<= 1) {
    if ((tid & (2 * s - 1)) == 0) red[tid] = aff_compose(red[tid], red[tid + s]);
    __syncthreads();
  }
  if (tid == 0)
    blocksum[blockIdx.x] = make_float4(red[0].aV, red[0].bV, red[0].aA, red[0].bA);
}

// ---------------- K2: suffix scan of block summaries -> per-block carries ----------------
__global__ __launch_bounds__(MAXBLK) void k2_scan(const float4* __restrict__ blocksum,
                                                  float2* __restrict__ carry, int nblk) {
  __shared__ float4 b0[MAXBLK];
  __shared__ float4 b1[MAXBLK];
  const int tid = threadIdx.x;
  if (nblk <= MAXBLK) {
    if (tid < nblk) b0[tid] = blocksum[tid];
    __syncthreads();
    float4* cur = b0; float4* nxt = b1;
    for (int off = 1; off < nblk; off <<= 1) {
      float4 val = (tid < nblk) ? cur[tid] : make_float4(1.f, 0.f, 1.f, 0.f);
      if (tid + off < nblk) {
        float4 rg = cur[tid + off];
        Aff L = {val.x, val.y, val.z, val.w};
        Aff R = {rg.x, rg.y, rg.z, rg.w};
        Aff C = aff_compose(L, R);
        val = make_float4(C.aV, C.bV, C.aA, C.bA);
      }
      __syncthreads();
      nxt[tid] = val;
      __syncthreads();
      float4* t2 = cur; cur = nxt; nxt = t2;
    }
    if (tid < nblk) {
      float2 c;
      if (tid + 1 < nblk) { float4 s = cur[tid + 1]; c = make_float2(s.y, s.w); }
      else                 c = make_float2(0.0f, 0.0f);   // x_init = 0 at far right
      carry[tid] = c;
    }
  } else if (tid == 0) {          // generic serial fallback
    float cV = 0.0f, cA = 0.0f;
    for (int i = nblk - 1; i >= 0; --i) {
      carry[i] = make_float2(cV, cA);
      float4 s = blocksum[i];
      cV = fmaf(s.x, cV, s.y);
      cA = fmaf(s.z, cA, s.w);
    }
  }
}

// ---------------- K3: apply carries, write v_target + raw A, per-block stats ----------------
__global__ __launch_bounds__(BT) void k3_apply(const float* __restrict__ r,
                                               const float* __restrict__ v,
                                               const int* __restrict__ m,
                                               const float2* __restrict__ carry,
                                               float* __restrict__ outA,
                                               float* __restrict__ outVT,
                                               double2* __restrict__ partial, int B) {
  __shared__ alignas(16) float lr[BT * TP];
  __shared__ alignas(16) float lv[BT * TP];
  __shared__ alignas(16) int   lm[BT * TP];
  __shared__ Aff               s0[BT];
  __shared__ Aff               s1[BT];
  __shared__ double2           wred[BT / 32];
  const int tid  = threadIdx.x;
  const int base = blockIdx.x * CHUNK;

  stage_tile(lr, lv, lm, r, v, m, base, B, tid);

  float vb;
  if (tid < BT - 1) vb = lv[(tid + 1) * TP];
  else { int g = base + CHUNK; vb = (g < B) ? v[g] : 0.0f; }

  // intra-block suffix scan of per-thread summaries
  s0[tid] = seg_summary(lr, lv, lm, vb, tid);
  __syncthreads();
  Aff* cur = s0; Aff* nxt = s1;
#pragma unroll
  for (int off = 1; off < BT; off <<= 1) {
    Aff val = cur[tid];
    if (tid + off < BT) val = aff_compose(val, cur[tid + off]);
    __syncthreads();
    nxt[tid] = val;
    __syncthreads();
    Aff* t2 = cur; cur = nxt; nxt = t2;
  }

  const float2 bc = carry[blockIdx.x];
  float cV, cA;
  if (tid < BT - 1) {
    Aff s = cur[tid + 1];
    cV = fmaf(s.aV, bc.x, s.bV);
    cA = fmaf(s.aA, bc.y, s.bA);
  } else { cV = bc.x; cA = bc.y; }

  // final serial walk with carries; write outputs + accumulate stats
  float xv = cV, xa = cA, vnext = vb;
  const int sb = tid * TP;
  double lsum = 0.0, lsq = 0.0;
#pragma unroll
  for (int k = T - 1; k >= 0; --k) {
    float rk = lr[sb + k], vk = lv[sb + k], mk = (float)lm[sb + k];
    float vt    = fmaf(GAMMA * mk, xv, rk);
    float delta = rk + GAMMA * vnext * mk - vk;
    float Av    = fmaf(GTAU * mk, xa, delta);
    int gi = base + tid * T + k;
    if (gi < B) {
      outVT[gi] = vt;
      outA[gi]  = Av;
      lsum += (double)Av;
      lsq  += (double)Av * (double)Av;
    }
    xv = vt; xa = Av;
    vnext = vk;
  }

  // deterministic wave32 + LDS reduction of (sum, sumsq)
#pragma unroll
  for (int o = 16; o > 0; o >>= 1) {
    lsum += __shfl_down(lsum, o);
    lsq  += __shfl_down(lsq, o);
  }
  if ((tid & 31) == 0) wred[tid >> 5] = make_double2(lsum, lsq);
  __syncthreads();
  if (tid == 0) {
    double a = 0.0, b = 0.0;
#pragma unroll
    for (int i = 0; i < BT / 32; ++i) { a += wred[i].x; b += wred[i].y; }
    partial[blockIdx.x] = make_double2(a, b);
  }
}

// ---------------- K4: reduce partials -> mean, 1/std (ddof=1) ----------------
__global__ __launch_bounds__(MAXBLK) void k4_stats(const double2* __restrict__ partial,
                                                   int nblk, int B,
                                                   float* __restrict__ stats) {
  __shared__ double ssum[MAXBLK];
  __shared__ double ssq[MAXBLK];
  const int tid = threadIdx.x;
  double a = 0.0, b = 0.0;
  for (int i = tid; i < nblk; i += MAXBLK) { a += partial[i].x; b += partial[i].y; }
  ssum[tid] = a; ssq[tid] = b;
  __syncthreads();
  for (int s = MAXBLK / 2; s > 0; s >>= 1) {
    if (tid < s) { ssum[tid] += ssum[tid + s]; ssq[tid] += ssq[tid + s]; }
    __syncthreads();
  }
  if (tid == 0) {
    double n    = (double)B;
    double mean = ssum[0] / n;
    double var  = (ssq[0] - ssum[0] * ssum[0] / n) / (n - 1.0);
    stats[0] = (float)mean;
    stats[1] = (float)(1.0 / sqrt(var));
  }
}

// ---------------- K5: normalize A in place ----------------
__global__ void k5_norm(float* __restrict__ outA, const float* __restrict__ stats, int B) {
  const float mean = stats[0];
  const float is   = stats[1];
  int i4 = (blockIdx.x * blockDim.x + threadIdx.x) * 4;
  if (i4 + 3 < B) {
    float4 x = *(const float4*)&outA[i4];
    x.x = (x.x - mean) * is;
    x.y = (x.y - mean) * is;
    x.z = (x.z - mean) * is;
    x.w = (x.w - mean) * is;
    *(float4*)&outA[i4] = x;
  } else {
    for (int k = i4; k < B; ++k) outA[k] = (outA[k] - mean) * is;
  }
}

extern "C" void kernel_launch(void* const* d_in, const int* in_sizes, int n_in,
                              void* d_out, int out_size, void* d_ws, size_t ws_size,
                              hipStream_t stream) {
  const int B = in_sizes[0];
  const float* r = (const float*)d_in[0];
  const float* v = (const float*)d_in[1];
  const int*   m = (const int*)d_in[2];
  float* outA  = (float*)d_out;        // first B: normalized A_sa
  float* outVT = outA + B;             // next B: v_target

  const int nblk = (B + CHUNK - 1) / CHUNK;

  // workspace layout (all fully rewritten each call; no cross-call state)
  char* w = (char*)d_ws;
  double2* partial  = (double2*)w;  w += (size_t)nblk * sizeof(double2);
  float4*  blocksum = (float4*)w;   w += (size_t)nblk * sizeof(float4);
  float2*  carry    = (float2*)w;   w += (size_t)nblk * sizeof(float2);
  float*   stats    = (float*)w;

  k1_summary<<<nblk, BT, 0, stream>>>(r, v, m, blocksum, B);
  k2_scan<<<1, MAXBLK, 0, stream>>>(blocksum, carry, nblk);
  k3_apply<<<nblk, BT, 0, stream>>>(r, v, m, carry, outA, outVT, partial, B);
  k4_stats<<<1, MAXBLK, 0, stream>>>(partial, nblk, B, stats);
  const int n5 = (B + 4 * 256 - 1) / (4 * 256);
  k5_norm<<<n5, 256, 0, stream>>>(outA, stats, B);
}